// BlellochScan_1967095021837
// MI455X (gfx1250) — compile-verified
//
#include <hip/hip_runtime.h>

typedef float v2f __attribute__((ext_vector_type(2)));
typedef float v8f __attribute__((ext_vector_type(8)));

#define B_   4
#define L_   8192
#define C_   1024           // D*N = 32*32, contiguous fastest axis
#define TL   256            // tile length along L
#define NT   (L_ / TL)      // 32 tiles per batch

// ---------------------------------------------------------------------------
// Phase A: per-tile, per-channel sums. Grid (NT, B_), 256 threads.
// Each thread owns 4 channels (float4) -> one block covers all 1024 channels.
// Coalesced: 256 threads * 16B = 4KB per row. Leaves input resident in L2.
// ---------------------------------------------------------------------------
__global__ __launch_bounds__(256) void tile_sum_kernel(const float* __restrict__ x,
                                                       float* __restrict__ sums) {
  const int t  = blockIdx.x;
  const int b  = blockIdx.y;
  const int c4 = threadIdx.x * 4;
  const float4* xr = (const float4*)(x + (size_t)(b * L_ + t * TL) * C_ + c4);
  float4 s = {0.f, 0.f, 0.f, 0.f};
  for (int l = 0; l < TL; ++l) {
    float4 v = xr[(size_t)l * (C_ / 4)];
    s.x += v.x; s.y += v.y; s.z += v.z; s.w += v.w;
  }
  *(float4*)(sums + (size_t)(b * NT + t) * C_ + c4) = s;
}

// ---------------------------------------------------------------------------
// Phase B: exclusive scan of the NT tile sums per (b, c) column, in place.
// 4096 independent length-32 scans; one thread each. Grid 16 x 256.
// ---------------------------------------------------------------------------
__global__ __launch_bounds__(256) void tile_scan_kernel(float* __restrict__ sums) {
  const int gid = blockIdx.x * blockDim.x + threadIdx.x;  // 0..4095
  const int b = gid >> 10;
  const int c = gid & 1023;
  float* p = sums + (size_t)b * NT * C_ + c;
  float run = 0.f;
  for (int t = 0; t < NT; ++t) {
    float v = p[(size_t)t * C_];
    p[(size_t)t * C_] = run;
    run += v;
  }
}

// ---------------------------------------------------------------------------
// Phase C: in-tile inclusive scan via lower-triangular WMMA.
// Grid (8, NT, B_), 256 threads = 8 waves. Each wave owns 16 channels and
// walks its tile in 16-row sub-tiles:  D = Tri(16x16) * X(16x16) + carry,
// built from 4 chained V_WMMA_F32_16X16X4_F32 (K=4 chunks of Tri / X).
//
// Layouts per CDNA5 ISA 7.12.2 (wave32):
//   A 16x4 f32 : all lanes M = lane&15; VGPRv holds K = 4k + v + 2*(lane>>4)
//   B  4x16 f32: all lanes N = lane&15; VGPRv holds K = 4k + v + 2*(lane>>4)
//   C/D 16x16  : VGPRr = row r (lanes 0-15) / row r+8 (lanes 16-31), N = lane&15
// New carry = D[15, c] lives in acc[7] of lanes 16..31 -> wave32 shuffle.
// ---------------------------------------------------------------------------
__global__ __launch_bounds__(256) void wmma_scan_kernel(const float* __restrict__ x,
                                                        const float* __restrict__ offs,
                                                        float* __restrict__ out) {
  const int wave = threadIdx.x >> 5;
  const int lane = threadIdx.x & 31;
  const int half = lane >> 4;       // 0 or 1
  const int n    = lane & 15;       // column within the 16-wide stripe
  const int cg   = blockIdx.x;      // channel group (0..7)
  const int t    = blockIdx.y;      // tile
  const int b    = blockIdx.z;      // batch
  const int col  = cg * 128 + wave * 16 + n;

  // Lower-triangular A chunks (inclusive scan): A[M,K] = (K <= M).
  v2f a[4];
#pragma unroll
  for (int k = 0; k < 4; ++k) {
    a[k][0] = (4 * k + 0 + 2 * half <= n) ? 1.0f : 0.0f;
    a[k][1] = (4 * k + 1 + 2 * half <= n) ? 1.0f : 0.0f;
  }

  float carry = offs[(size_t)(b * NT + t) * C_ + col];

  const float* xp = x   + (size_t)(b * L_ + t * TL) * C_ + col;
  float*       op = out + (size_t)(b * L_ + t * TL) * C_ + col;

  for (int s = 0; s < TL / 16; ++s) {
    const float* xs = xp + (size_t)(s * 16) * C_;
    float*       os = op + (size_t)(s * 16) * C_;

    v8f acc;
#pragma unroll
    for (int r = 0; r < 8; ++r) acc[r] = carry;

#pragma unroll
    for (int k = 0; k < 4; ++k) {
      v2f bm;
      bm[0] = xs[(size_t)(4 * k + 0 + 2 * half) * C_];
      bm[1] = xs[(size_t)(4 * k + 1 + 2 * half) * C_];
      acc = __builtin_amdgcn_wmma_f32_16x16x4_f32(false, a[k], false, bm,
                                                  (short)0, acc,
                                                  false, false);
    }

#pragma unroll
    for (int r = 0; r < 8; ++r)
      os[(size_t)(r + 8 * half) * C_] = acc[r];

    // Broadcast D[15, col] (held by lane 16+n in acc[7]) to the whole column.
    carry = __shfl(acc[7], 16 + n, 32);
  }
}

// ---------------------------------------------------------------------------
extern "C" void kernel_launch(void* const* d_in, const int* in_sizes, int n_in,
                              void* d_out, int out_size, void* d_ws, size_t ws_size,
                              hipStream_t stream) {
  (void)in_sizes; (void)n_in; (void)out_size; (void)ws_size;
  const float* x   = (const float*)d_in[0];
  float*       out = (float*)d_out;
  float*       sums = (float*)d_ws;    // B_*NT*C_ floats = 512 KB scratch

  tile_sum_kernel <<<dim3(NT, B_), 256, 0, stream>>>(x, sums);
  tile_scan_kernel<<<16, 256, 0, stream>>>(sums);
  wmma_scan_kernel<<<dim3(8, NT, B_), 256, 0, stream>>>(x, sums, out);
}